// GCN_27101243638257
// MI455X (gfx1250) — compile-verified
//
#include <hip/hip_runtime.h>
#include <hip/hip_bf16.h>

#define NNODES 100000
#define NEDGES 1600000
#define FDIM   128
#define FOUT   64
#define BN_EPS 1e-5f

typedef __attribute__((ext_vector_type(2))) float v2f;
typedef __attribute__((ext_vector_type(8))) float v8f;

__device__ __forceinline__ void atomic_add_f32(float* p, float v) {
    // gfx1250 has native GLOBAL_ATOMIC_ADD_F32; unsafeAtomicAdd lowers to it.
    unsafeAtomicAdd(p, v);
}

// ---------------------------------------------------------------- degrees ---
__global__ void gcn_deg_kernel(const int* __restrict__ src, const int* __restrict__ dst,
                               float* __restrict__ deg_out, float* __restrict__ deg_in) {
    int e = blockIdx.x * blockDim.x + threadIdx.x;
    if (e < NEDGES) {
        atomic_add_f32(&deg_out[src[e]], 1.0f);
        atomic_add_f32(&deg_in[dst[e]], 1.0f);
    }
}

__global__ void gcn_rsqrt_kernel(float* __restrict__ deg_out, float* __restrict__ deg_in) {
    int i = blockIdx.x * blockDim.x + threadIdx.x;
    if (i < NNODES) {
        float a = deg_out[i]; a = a < 1.0f ? 1.0f : a;
        deg_out[i] = 1.0f / sqrtf(a);
        float b = deg_in[i];  b = b < 1.0f ? 1.0f : b;
        deg_in[i] = 1.0f / sqrtf(b);
    }
}

// -------------------------------------------------- GEMM: (H*rs_out) @ W ----
// One wave computes one 16x16 C tile via V_WMMA_F32_16X16X4_F32 (K-loop of 32).
// NOUT/16 waves per block cover the full output width for a 16-row tile.
template <int NOUT>
__launch_bounds__((NOUT / 16) * 32)
__global__ void gcn_gemm_wmma(const float* __restrict__ H, const float* __restrict__ rs,
                              const float* __restrict__ W, float* __restrict__ out) {
    constexpr int K = FDIM;
    constexpr int NTHR = (NOUT / 16) * 32;
    __shared__ float Alds[16][K + 4];   // pad 128->132: conflict-free column reads

    const int tid = threadIdx.x;
    const int r0  = blockIdx.x * 16;    // 6250 * 16 == 100000 exactly

    // Cooperative load of 16x128 A tile, fused with rs_out row scaling.
    for (int idx = tid; idx < 16 * K; idx += NTHR) {
        int row = idx >> 7;
        int col = idx & (K - 1);
        Alds[row][col] = H[(r0 + row) * K + col] * rs[r0 + row];
    }
    __syncthreads();

    const int lane = tid & 31;
    const int wave = tid >> 5;
    const int n0   = wave * 16;
    const int half = lane >> 4;        // 0: K+0/K+1, 1: K+2/K+3 (ISA A 16x4 layout)
    const int l16  = lane & 15;
    const int koff = half * 2;

    v8f c = {};
    #pragma unroll 4
    for (int k = 0; k < K; k += 4) {
        v2f a, b;
        a.x = Alds[l16][k + koff];            // A: M=l16, K=k+koff
        a.y = Alds[l16][k + koff + 1];        //    M=l16, K=k+koff+1
        const float* wp = &W[(k + koff) * NOUT + n0 + l16];
        b.x = wp[0];                          // B: K=k+koff,   N=n0+l16
        b.y = wp[NOUT];                       //    K=k+koff+1, N=n0+l16
        c = __builtin_amdgcn_wmma_f32_16x16x4_f32(
                false, a, false, b, (short)0, c, false, false);
    }

    // C/D layout: VGPR i -> M = i (lanes 0-15) or i+8 (lanes 16-31), N = lane%16
    float* op = &out[(r0 + half * 8) * NOUT + n0 + l16];
    #pragma unroll
    for (int i = 0; i < 8; ++i)
        op[i * NOUT] = c[i];
}

// ------------------------------------------------- SpMM edge scatter-add ----
template <int F>
__global__ void gcn_spmm_scatter(const float* __restrict__ Hin, const int* __restrict__ src,
                                 const int* __restrict__ dst, float* __restrict__ out) {
    constexpr int TPE = F / 4;  // threads per edge, float4 per thread
    long long t = (long long)blockIdx.x * blockDim.x + threadIdx.x;
    if (t >= (long long)NEDGES * TPE) return;
    int e = (int)(t / TPE);
    int c = (int)(t % TPE) * 4;
    int s = src[e], d = dst[e];
    const float4 v = *reinterpret_cast<const float4*>(&Hin[s * F + c]);
    float* o = &out[d * F + c];
    atomic_add_f32(o + 0, v.x);
    atomic_add_f32(o + 1, v.y);
    atomic_add_f32(o + 2, v.z);
    atomic_add_f32(o + 3, v.w);
}

// ------------------------- agg*rs_in + bias (in place) + batch statistics ---
__global__ __launch_bounds__(256)
void gcn_scale_bias_stats(float* __restrict__ h, const float* __restrict__ rs,
                          const float* __restrict__ bias, float* __restrict__ stats) {
    __shared__ float ssum[256];
    __shared__ float ssq[256];
    const int tid = threadIdx.x;
    const int f   = tid & (FDIM - 1);
    const int rib = tid >> 7;           // 0 or 1: two rows per block sweep
    const float bf = bias[f];
    float bsum = 0.f, bsq = 0.f;
    for (int row = blockIdx.x * 2 + rib; row < NNODES; row += gridDim.x * 2) {
        float v = h[row * FDIM + f] * rs[row] + bf;
        h[row * FDIM + f] = v;
        bsum += v;
        bsq  += v * v;
    }
    ssum[tid] = bsum; ssq[tid] = bsq;
    __syncthreads();
    if (tid < FDIM) {
        atomic_add_f32(&stats[f],        ssum[tid] + ssum[tid + FDIM]);
        atomic_add_f32(&stats[FDIM + f], ssq[tid]  + ssq[tid + FDIM]);
    }
}

__global__ void gcn_bn_relu(float* __restrict__ h, const float* __restrict__ stats,
                            const float* __restrict__ gamma, const float* __restrict__ beta) {
    int i = blockIdx.x * blockDim.x + threadIdx.x;
    if (i >= NNODES * FDIM) return;
    int f = i & (FDIM - 1);
    const float invN = 1.0f / (float)NNODES;
    float mean = stats[f] * invN;
    float var  = stats[FDIM + f] * invN - mean * mean;
    float inv  = 1.0f / sqrtf(var + BN_EPS);
    float v = (h[i] - mean) * inv * gamma[f] + beta[f];
    h[i] = v > 0.f ? v : 0.f;
}

__global__ void gcn_final_scale(float* __restrict__ out, const float* __restrict__ rs,
                                const float* __restrict__ b2) {
    int i = blockIdx.x * blockDim.x + threadIdx.x;
    if (i >= NNODES * FOUT) return;
    int row = i >> 6;
    int f   = i & (FOUT - 1);
    out[i] = out[i] * rs[row] + b2[f];
}

// ----------------------------------------------------------------------------
extern "C" void kernel_launch(void* const* d_in, const int* in_sizes, int n_in,
                              void* d_out, int out_size, void* d_ws, size_t ws_size,
                              hipStream_t stream) {
    const float* in_feat = (const float*)d_in[0];
    const int*   src     = (const int*)d_in[1];
    const int*   dst     = (const int*)d_in[2];
    const float* W0 = (const float*)d_in[3];
    const float* b0 = (const float*)d_in[4];
    const float* W1 = (const float*)d_in[5];
    const float* b1 = (const float*)d_in[6];
    const float* W2 = (const float*)d_in[7];
    const float* b2 = (const float*)d_in[8];
    const float* g0 = (const float*)d_in[9];
    const float* be0 = (const float*)d_in[10];
    const float* g1 = (const float*)d_in[11];
    const float* be1 = (const float*)d_in[12];
    float* outF = (float*)d_out;

    char* ws = (char*)d_ws;
    size_t off = 0;
    auto alloc = [&](size_t bytes) -> void* {
        void* p = ws + off;
        off = (off + bytes + 255) & ~(size_t)255;
        return p;
    };
    float* rs_out = (float*)alloc((size_t)NNODES * 4);
    float* rs_in  = (float*)alloc((size_t)NNODES * 4);
    float* bufA   = (float*)alloc((size_t)NNODES * FDIM * 4);
    float* bufB   = (float*)alloc((size_t)NNODES * FDIM * 4);
    float* stats  = (float*)alloc(2 * FDIM * 4);

    const int B = 256;
    const int degBlocks  = (NEDGES + B - 1) / B;
    const int nodeBlocks = (NNODES + B - 1) / B;

    // ---- degrees -> rsqrt scales
    hipMemsetAsync(rs_out, 0, (size_t)NNODES * 4, stream);
    hipMemsetAsync(rs_in,  0, (size_t)NNODES * 4, stream);
    gcn_deg_kernel<<<degBlocks, B, 0, stream>>>(src, dst, rs_out, rs_in);
    gcn_rsqrt_kernel<<<nodeBlocks, B, 0, stream>>>(rs_out, rs_in);

    const int gemmBlocks  = NNODES / 16;                       // 6250
    const long long e128  = (long long)NEDGES * (FDIM / 4);
    const int spmm128Blks = (int)((e128 + B - 1) / B);
    const long long e64   = (long long)NEDGES * (FOUT / 4);
    const int spmm64Blks  = (int)((e64 + B - 1) / B);
    const int bnBlocks    = (NNODES * FDIM + B - 1) / B;

    // ---- layer 0: in_feat -> bufB (post BN+ReLU)
    gcn_gemm_wmma<FDIM><<<gemmBlocks, (FDIM / 16) * 32, 0, stream>>>(in_feat, rs_out, W0, bufA);
    hipMemsetAsync(bufB, 0, (size_t)NNODES * FDIM * 4, stream);
    gcn_spmm_scatter<FDIM><<<spmm128Blks, B, 0, stream>>>(bufA, src, dst, bufB);
    hipMemsetAsync(stats, 0, 2 * FDIM * 4, stream);
    gcn_scale_bias_stats<<<1024, 256, 0, stream>>>(bufB, rs_in, b0, stats);
    gcn_bn_relu<<<bnBlocks, B, 0, stream>>>(bufB, stats, g0, be0);

    // ---- layer 1: bufB -> bufB
    gcn_gemm_wmma<FDIM><<<gemmBlocks, (FDIM / 16) * 32, 0, stream>>>(bufB, rs_out, W1, bufA);
    hipMemsetAsync(bufB, 0, (size_t)NNODES * FDIM * 4, stream);
    gcn_spmm_scatter<FDIM><<<spmm128Blks, B, 0, stream>>>(bufA, src, dst, bufB);
    hipMemsetAsync(stats, 0, 2 * FDIM * 4, stream);
    gcn_scale_bias_stats<<<1024, 256, 0, stream>>>(bufB, rs_in, b1, stats);
    gcn_bn_relu<<<bnBlocks, B, 0, stream>>>(bufB, stats, g1, be1);

    // ---- layer 2: bufB -> d_out (no BN/ReLU)
    gcn_gemm_wmma<FOUT><<<gemmBlocks, (FOUT / 16) * 32, 0, stream>>>(bufB, rs_out, W2, bufA);
    hipMemsetAsync(outF, 0, (size_t)NNODES * FOUT * 4, stream);
    gcn_spmm_scatter<FOUT><<<spmm64Blks, B, 0, stream>>>(bufA, src, dst, outF);
    gcn_final_scale<<<(NNODES * FOUT + B - 1) / B, B, 0, stream>>>(outF, rs_in, b2);
}